// GaussianMoELayer_5592047419818
// MI455X (gfx1250) — compile-verified
//
#include <hip/hip_runtime.h>
#include <hip/hip_bf16.h>

// Problem constants (fixed by the reference setup)
#define NTOK 4096   // B*T = 2*2048
#define DDIM 512
#define HDIM 2048
#define NEXP 8

#define HALF_LOG_2PI 0.91893853320467274178f

// d_out layout (flat element offsets, fp32/int32 elements)
#define Y_OFF   ((size_t)0)                          // N*D floats
#define LP_OFF  ((size_t)NTOK * DDIM)                // N*E floats
#define W_OFF   (LP_OFF + (size_t)NTOK * NEXP)       // N*2 floats
#define IDX_OFF (W_OFF + (size_t)NTOK * 2)           // N*2 ints

typedef float v2f __attribute__((ext_vector_type(2)));
typedef float v8f __attribute__((ext_vector_type(8)));

#if __has_builtin(__builtin_amdgcn_global_load_async_to_lds_b128)
#define HAVE_ASYNC_LDS 1
// Exact parameter types per hipcc diagnostic: vector_size(16) int pointers.
typedef int v4i_vs __attribute__((vector_size(16)));
typedef v4i_vs __attribute__((address_space(1)))* async_gptr;
typedef v4i_vs __attribute__((address_space(3)))* async_lptr;
#endif

// ---------------------------------------------------------------------------
// Kernel 1: Gaussian routing + top-2 + softmax + per-expert token lists
// ---------------------------------------------------------------------------
__global__ __launch_bounds__(64)
void moe_route_kernel(const float* __restrict__ x,
                      const float* __restrict__ mus,
                      const float* __restrict__ lsg,
                      float* __restrict__ out,
                      int* __restrict__ cnt,
                      int* __restrict__ perm,
                      float* __restrict__ pw) {
  const int n   = blockIdx.x;
  const int tid = threadIdx.x;

  __shared__ float red[64];
  __shared__ float lp[NEXP];

  float acc[NEXP];
#pragma unroll
  for (int e = 0; e < NEXP; ++e) acc[e] = 0.0f;

  for (int d = tid; d < DDIM; d += 64) {
    const float xv = x[(size_t)n * DDIM + d];
#pragma unroll
    for (int e = 0; e < NEXP; ++e) {
      const float l   = lsg[e * DDIM + d];
      const float inv = expf(-l);
      const float z   = (xv - mus[e * DDIM + d]) * inv;
      acc[e] += -0.5f * z * z - l - HALF_LOG_2PI;
    }
  }

  for (int e = 0; e < NEXP; ++e) {
    red[tid] = acc[e];
    __syncthreads();
    for (int s = 32; s > 0; s >>= 1) {
      if (tid < s) red[tid] += red[tid + s];
      __syncthreads();
    }
    if (tid == 0) lp[e] = red[0];
    __syncthreads();
  }

  if (tid == 0) {
#pragma unroll
    for (int e = 0; e < NEXP; ++e) out[LP_OFF + (size_t)n * NEXP + e] = lp[e];

    // top-2, ties broken by lowest index (matches jax.lax.top_k)
    int i0 = 0; float v0 = lp[0];
#pragma unroll
    for (int e = 1; e < NEXP; ++e) { if (lp[e] > v0) { v0 = lp[e]; i0 = e; } }
    int i1 = -1; float v1 = -3.4e38f;
#pragma unroll
    for (int e = 0; e < NEXP; ++e) {
      if (e != i0 && lp[e] > v1) { v1 = lp[e]; i1 = e; }
    }

    const float ew = expf(v1 - v0);      // v1 <= v0
    const float w0 = 1.0f / (1.0f + ew);
    const float w1 = ew / (1.0f + ew);

    out[W_OFF + (size_t)n * 2 + 0] = w0;
    out[W_OFF + (size_t)n * 2 + 1] = w1;
    int* idxo = (int*)out;
    idxo[IDX_OFF + (size_t)n * 2 + 0] = i0;
    idxo[IDX_OFF + (size_t)n * 2 + 1] = i1;

    int p0 = atomicAdd(&cnt[i0], 1);
    perm[i0 * NTOK + p0] = n;  pw[i0 * NTOK + p0] = w0;
    int p1 = atomicAdd(&cnt[i1], 1);
    perm[i1 * NTOK + p1] = n;  pw[i1 * NTOK + p1] = w1;
  }
}

// ---------------------------------------------------------------------------
// Kernel 2: fused gather -> GEMM1(WMMA f32) -> GELU -> GEMM2(WMMA f32)
//           -> weighted atomic scatter.  h tile (16 x 2048 fp32 = 128 KB)
//           lives entirely in the 320 KB WGP LDS.  Each wave computes two
//           adjacent 16x16 N-tiles so one A fragment feeds two WMMAs.
// ---------------------------------------------------------------------------
__global__ __launch_bounds__(256)
void moe_mlp_kernel(const float* __restrict__ x,
                    const float* __restrict__ W1,
                    const float* __restrict__ b1,
                    const float* __restrict__ W2,
                    const float* __restrict__ b2,
                    const int* __restrict__ cnt,
                    const int* __restrict__ perm,
                    const float* __restrict__ pw,
                    float* __restrict__ y) {
  extern __shared__ float smem[];
  float* xg = smem;                 // 16 * 512  fp32 (32 KB)
  float* ht = smem + 16 * DDIM;     // 16 * 2048 fp32 (128 KB)
  __shared__ int   tokS[16];
  __shared__ float pwS[16];

  const int e      = blockIdx.y;
  const int tile   = blockIdx.x;
  const int mcount = cnt[e];
  if (tile * 16 >= mcount) return;  // uniform across block

  const int tid = threadIdx.x;
  if (tid < 16) {
    const int s = tile * 16 + tid;
    const int t = (s < mcount) ? perm[e * NTOK + s] : -1;
    tokS[tid] = t;
    pwS[tid]  = (s < mcount) ? pw[e * NTOK + s] : 0.0f;
  }
  __syncthreads();

  // gather 16 token rows of x into LDS (pad rows -> zero), 16B granules.
  // Prefer the gfx1250 async global->LDS path (ASYNCcnt-tracked DMA).
  for (int i4 = tid; i4 < 16 * DDIM / 4; i4 += 256) {
    const int s  = i4 >> 7;            // / (512/4)
    const int d4 = i4 & 127;
    const int t  = tokS[s];
#ifdef HAVE_ASYNC_LDS
    if (t >= 0) {
      __builtin_amdgcn_global_load_async_to_lds_b128(
          (async_gptr)(x + (size_t)t * DDIM + (size_t)d4 * 4),
          (async_lptr)(xg + (size_t)i4 * 4),
          0, 0);
    } else {
      ((float4*)xg)[i4] = make_float4(0.f, 0.f, 0.f, 0.f);
    }
#else
    float4 v = (t >= 0) ? ((const float4*)(x + (size_t)t * DDIM))[d4]
                        : make_float4(0.f, 0.f, 0.f, 0.f);
    ((float4*)xg)[i4] = v;
#endif
  }
#ifdef HAVE_ASYNC_LDS
#if __has_builtin(__builtin_amdgcn_s_wait_asynccnt)
  __builtin_amdgcn_s_wait_asynccnt(0);
#else
  asm volatile("s_wait_asynccnt 0x0" ::: "memory");
#endif
#endif
  __syncthreads();

  const int wave = tid >> 5;
  const int lane = tid & 31;
  const int mrow = lane & 15;            // A-fragment row
  const int koff = (lane >> 4) * 2;      // A/B K sub-offset per lane half
  const int ncol = lane & 15;            // B/C/D column

  // ---- GEMM1: h = gelu(xg @ W1[e] + b1[e]) --------------------------------
  const float* W1e = W1 + (size_t)e * DDIM * HDIM;
  for (int hp = wave; hp < HDIM / 32; hp += 8) {   // 64 pairs of N-tiles
    const int hbase = hp * 32;
    v8f c0 = {};
    v8f c1 = {};
    const float* Bp0 = W1e + hbase + ncol;
    const float* Bp1 = Bp0 + 16;
#pragma unroll 8
    for (int kk = 0; kk < DDIM; kk += 4) {
      if ((kk & 12) == 0)  // one prefetch per 16 K-values, ~128 rows ahead
        __builtin_prefetch(Bp0 + (size_t)((kk + 128) & (DDIM - 1)) * HDIM, 0, 0);
      v2f a = *(const v2f*)&xg[mrow * DDIM + kk + koff];
      v2f b0, b1v;
      b0.x  = Bp0[(size_t)(kk + koff)     * HDIM];
      b0.y  = Bp0[(size_t)(kk + koff + 1) * HDIM];
      b1v.x = Bp1[(size_t)(kk + koff)     * HDIM];
      b1v.y = Bp1[(size_t)(kk + koff + 1) * HDIM];
      c0 = __builtin_amdgcn_wmma_f32_16x16x4_f32(false, a, false, b0,
                                                 (short)0, c0, false, false);
      c1 = __builtin_amdgcn_wmma_f32_16x16x4_f32(false, a, false, b1v,
                                                 (short)0, c1, false, false);
    }
    const float bias0 = b1[e * HDIM + hbase + ncol];
    const float bias1 = b1[e * HDIM + hbase + 16 + ncol];
#pragma unroll
    for (int i = 0; i < 8; ++i) {
      const int m = i + 8 * (lane >> 4);
      float v0 = c0[i] + bias0;
      float v1 = c1[i] + bias1;
      v0 = 0.5f * v0 * (1.0f + erff(v0 * 0.70710678118654752f));  // exact GELU
      v1 = 0.5f * v1 * (1.0f + erff(v1 * 0.70710678118654752f));
      ht[m * HDIM + hbase + ncol]      = v0;
      ht[m * HDIM + hbase + 16 + ncol] = v1;
    }
  }
  __syncthreads();

  // ---- GEMM2: out = h @ W2[e] + b2[e]; scatter w * out into y -------------
  const float* W2e = W2 + (size_t)e * HDIM * DDIM;
  for (int dp = wave; dp < DDIM / 32; dp += 8) {   // 16 pairs of N-tiles
    const int dbase = dp * 32;
    v8f c0 = {};
    v8f c1 = {};
    const float* Bp0 = W2e + dbase + ncol;
    const float* Bp1 = Bp0 + 16;
#pragma unroll 8
    for (int kk = 0; kk < HDIM; kk += 4) {
      if ((kk & 12) == 0)
        __builtin_prefetch(Bp0 + (size_t)((kk + 128) & (HDIM - 1)) * DDIM, 0, 0);
      v2f a = *(const v2f*)&ht[mrow * HDIM + kk + koff];
      v2f b0, b1v;
      b0.x  = Bp0[(size_t)(kk + koff)     * DDIM];
      b0.y  = Bp0[(size_t)(kk + koff + 1) * DDIM];
      b1v.x = Bp1[(size_t)(kk + koff)     * DDIM];
      b1v.y = Bp1[(size_t)(kk + koff + 1) * DDIM];
      c0 = __builtin_amdgcn_wmma_f32_16x16x4_f32(false, a, false, b0,
                                                 (short)0, c0, false, false);
      c1 = __builtin_amdgcn_wmma_f32_16x16x4_f32(false, a, false, b1v,
                                                 (short)0, c1, false, false);
    }
    const float bias0 = b2[e * DDIM + dbase + ncol];
    const float bias1 = b2[e * DDIM + dbase + 16 + ncol];
#pragma unroll
    for (int i = 0; i < 8; ++i) {
      const int m = i + 8 * (lane >> 4);
      const int t = tokS[m];
      if (t >= 0) {
        const float w = pwS[m];
        atomicAdd(&y[(size_t)t * DDIM + dbase + ncol],      (c0[i] + bias0) * w);
        atomicAdd(&y[(size_t)t * DDIM + dbase + 16 + ncol], (c1[i] + bias1) * w);
      }
    }
  }
}

// ---------------------------------------------------------------------------
extern "C" void kernel_launch(void* const* d_in, const int* in_sizes, int n_in,
                              void* d_out, int out_size, void* d_ws, size_t ws_size,
                              hipStream_t stream) {
  (void)in_sizes; (void)n_in; (void)out_size; (void)ws_size;

  const float* x   = (const float*)d_in[0];
  const float* mus = (const float*)d_in[1];
  const float* lsg = (const float*)d_in[2];
  const float* W1  = (const float*)d_in[3];
  const float* b1  = (const float*)d_in[4];
  const float* W2  = (const float*)d_in[5];
  const float* b2  = (const float*)d_in[6];
  float* out = (float*)d_out;

  // workspace: [cnt: E ints][perm: E*N ints][pw: E*N floats]
  int*   cnt  = (int*)d_ws;
  int*   perm = (int*)((char*)d_ws + 256);
  float* pw   = (float*)((char*)d_ws + 256 + (size_t)NEXP * NTOK * sizeof(int));

  (void)hipMemsetAsync(cnt, 0, NEXP * sizeof(int), stream);
  (void)hipMemsetAsync(out, 0, (size_t)NTOK * DDIM * sizeof(float), stream);

  moe_route_kernel<<<NTOK, 64, 0, stream>>>(x, mus, lsg, out, cnt, perm, pw);

  const size_t ldsBytes = (size_t)(16 * DDIM + 16 * HDIM) * sizeof(float); // 160 KB
  (void)hipFuncSetAttribute(reinterpret_cast<const void*>(moe_mlp_kernel),
                            hipFuncAttributeMaxDynamicSharedMemorySize,
                            (int)ldsBytes);

  dim3 grid(NTOK / 16, NEXP);  // worst case: all tokens on one expert
  moe_mlp_kernel<<<grid, 256, ldsBytes, stream>>>(x, W1, b1, W2, b2,
                                                  cnt, perm, pw, out);
}